// MultiChev1_26645977104435
// MI455X (gfx1250) — compile-verified
//
#include <hip/hip_runtime.h>
#include <hip/hip_bf16.h>

typedef __attribute__((ext_vector_type(2))) float v2f;
typedef __attribute__((ext_vector_type(8))) float v8f;

#define IN_CH 128
#define NW 6                      // number of weight matrices packed together

// ---------------------------------------------------------------- zero scratch
__global__ void mc_zero_f32(float* __restrict__ p, long n) {
    long i = (long)blockIdx.x * blockDim.x + threadIdx.x;
    long stride = (long)gridDim.x * blockDim.x;
    for (; i < n; i += stride) p[i] = 0.0f;
}

// ---------------------------------------------------------------- deg[src] += w
__global__ void mc_deg(const int* __restrict__ src, const float* __restrict__ w,
                       float* __restrict__ deg, int E) {
    int e = blockIdx.x * blockDim.x + threadIdx.x;
    if (e < E) unsafeAtomicAdd(&deg[src[e]], w[e]);
}

// ---------------------------------------------------------------- deg -> rsqrt(deg)
__global__ void mc_dis(float* __restrict__ deg, int n) {
    int i = blockIdx.x * blockDim.x + threadIdx.x;
    if (i < n) {
        float d = deg[i];
        deg[i] = (d > 0.0f) ? rsqrtf(d) : 0.0f;
    }
}

// ---------------------------------------------------------------- per-edge norm
__global__ void mc_norm(const int* __restrict__ src, const int* __restrict__ dst,
                        const float* __restrict__ w, const float* __restrict__ dis,
                        float* __restrict__ nrm, int E) {
    int e = blockIdx.x * blockDim.x + threadIdx.x;
    if (e < E) nrm[e] = -dis[src[e]] * w[e] * dis[dst[e]];
}

// ---------------------------------------------------------------- SpMM scatter:
// out[dst*ostride] += norm * h[src*hstride], one wave32 per edge,
// float4 per lane (32*4 = 128 channels).
__global__ void mc_prop(const float* __restrict__ h, long hstride,
                        const int* __restrict__ src, const int* __restrict__ dst,
                        const float* __restrict__ nrm,
                        float* __restrict__ out, long ostride, int E) {
    int e = blockIdx.x * (blockDim.x >> 5) + (threadIdx.x >> 5);
    if (e >= E) return;
    int lane = threadIdx.x & 31;
    float nv = nrm[e];
    int s = src[e];
    int d = dst[e];
    const float4* hp = (const float4*)(h + (long)s * hstride);
    float4 v = hp[lane];
    float* op = out + (long)d * ostride + lane * 4;
    unsafeAtomicAdd(op + 0, nv * v.x);
    unsafeAtomicAdd(op + 1, nv * v.y);
    unsafeAtomicAdd(op + 2, nv * v.z);
    unsafeAtomicAdd(op + 3, nv * v.w);
}

// ---------------------------------------------------------------- weight pack:
// All six weights transposed + zero-padded + interleaved into ONE buffer:
//   wt[(c*6 + wi)*128 + k] = (c < OUT) ? w_wi[k*OUT + c] : 0
// so the GEMM addresses every B fragment from a single base pointer with
// small immediate offsets.
__global__ void mc_pack6(const float* __restrict__ w0, const float* __restrict__ w1,
                         const float* __restrict__ w2, const float* __restrict__ w3,
                         const float* __restrict__ w4, const float* __restrict__ w5,
                         float* __restrict__ wt, int OUT, int OUT_PAD) {
    int i = blockIdx.x * blockDim.x + threadIdx.x;
    if (i >= OUT_PAD * NW * IN_CH) return;
    int k  = i & (IN_CH - 1);
    int t  = i >> 7;          // c*6 + wi
    int wi = t % NW;
    int c  = t / NW;
    const float* w = (wi == 0) ? w0 : (wi == 1) ? w1 : (wi == 2) ? w2
                   : (wi == 3) ? w3 : (wi == 4) ? w4 : w5;
    wt[i] = (c < OUT) ? w[(long)k * OUT + c] : 0.0f;
}

// ---------------------------------------------------------------- fused 3-scale GEMM
// One wave32 owns 16 rows x 32 cols (two 16x16 WMMA subtiles), fp32 WMMA
// 16x16x4 over K=128. Three base pointers total:
//   x   : tx0 fragments
//   pr  : tx1 at +0, p2 at +128 floats (interleaved rows, +512B immediate)
//   wb  : all 12 B fragments at wi*512B / +49152B immediates
__global__ void __launch_bounds__(128)
mc_gemm(const float* __restrict__ x, const float* __restrict__ pbuf,
        const float* __restrict__ wt,
        const float* __restrict__ b1, const float* __restrict__ b2,
        const float* __restrict__ b3,
        float* __restrict__ out, int Nn, int OUT, int ntp) {
    const int mtiles = (Nn + 15) >> 4;
    int tile = blockIdx.x * (blockDim.x >> 5) + (threadIdx.x >> 5);
    if (tile >= mtiles * ntp) return;    // whole-wave uniform exit

    int mt = tile / ntp;
    int nt = tile - mt * ntp;
    int lane = threadIdx.x & 31;
    int l16  = lane & 15;
    int half = lane >> 4;
    int koff = half * 2;

    int arow = mt * 16 + l16;
    if (arow >= Nn) arow = Nn - 1;       // safe clamp (N%16==0 in practice)
    int colA = nt * 32 + l16;            // first 16x16 subtile column
    int colB = colA + 16;                // second subtile column

    const float* xr = x    + (long)arow * IN_CH + koff;
    const float* pr = pbuf + (long)arow * (2 * IN_CH) + koff;       // tx1 | p2
    const float* wb = wt   + ((long)colA * NW) * IN_CH + koff;      // packed B

    const int P2  = IN_CH;               // p2 immediate (floats)
    const int CB  = 16 * NW * IN_CH;     // colB immediate (floats) = 12288

    v8f acc1A = {}, acc2A = {}, acc3A = {};
    v8f acc1B = {}, acc2B = {}, acc3B = {};

#pragma unroll 4
    for (int k0 = 0; k0 < IN_CH; k0 += 4) {
        v2f a0 = *(const v2f*)(xr + k0);
        v2f a1 = *(const v2f*)(pr + k0);
        v2f ap = *(const v2f*)(pr + k0 + P2);
        v2f a2 = 2.0f * ap - a0;          // tx2 = 2*prop(tx1) - tx0

        v2f b10A = *(const v2f*)(wb + k0 + 0 * IN_CH);
        v2f b20A = *(const v2f*)(wb + k0 + 1 * IN_CH);
        v2f b21A = *(const v2f*)(wb + k0 + 2 * IN_CH);
        v2f b30A = *(const v2f*)(wb + k0 + 3 * IN_CH);
        v2f b31A = *(const v2f*)(wb + k0 + 4 * IN_CH);
        v2f b32A = *(const v2f*)(wb + k0 + 5 * IN_CH);
        v2f b10B = *(const v2f*)(wb + k0 + 0 * IN_CH + CB);
        v2f b20B = *(const v2f*)(wb + k0 + 1 * IN_CH + CB);
        v2f b21B = *(const v2f*)(wb + k0 + 2 * IN_CH + CB);
        v2f b30B = *(const v2f*)(wb + k0 + 3 * IN_CH + CB);
        v2f b31B = *(const v2f*)(wb + k0 + 4 * IN_CH + CB);
        v2f b32B = *(const v2f*)(wb + k0 + 5 * IN_CH + CB);

        acc1A = __builtin_amdgcn_wmma_f32_16x16x4_f32(false, a0, false, b10A, (short)0, acc1A, false, false);
        acc1B = __builtin_amdgcn_wmma_f32_16x16x4_f32(false, a0, false, b10B, (short)0, acc1B, false, false);
        acc2A = __builtin_amdgcn_wmma_f32_16x16x4_f32(false, a0, false, b20A, (short)0, acc2A, false, false);
        acc2B = __builtin_amdgcn_wmma_f32_16x16x4_f32(false, a0, false, b20B, (short)0, acc2B, false, false);
        acc2A = __builtin_amdgcn_wmma_f32_16x16x4_f32(false, a1, false, b21A, (short)0, acc2A, false, false);
        acc2B = __builtin_amdgcn_wmma_f32_16x16x4_f32(false, a1, false, b21B, (short)0, acc2B, false, false);
        acc3A = __builtin_amdgcn_wmma_f32_16x16x4_f32(false, a0, false, b30A, (short)0, acc3A, false, false);
        acc3B = __builtin_amdgcn_wmma_f32_16x16x4_f32(false, a0, false, b30B, (short)0, acc3B, false, false);
        acc3A = __builtin_amdgcn_wmma_f32_16x16x4_f32(false, a1, false, b31A, (short)0, acc3A, false, false);
        acc3B = __builtin_amdgcn_wmma_f32_16x16x4_f32(false, a1, false, b31B, (short)0, acc3B, false, false);
        acc3A = __builtin_amdgcn_wmma_f32_16x16x4_f32(false, a2, false, b32A, (short)0, acc3A, false, false);
        acc3B = __builtin_amdgcn_wmma_f32_16x16x4_f32(false, a2, false, b32B, (short)0, acc3B, false, false);
    }

    // Store: acc element r -> row mt*16 + r + 8*half, cols colA / colB.
    bool okA = colA < OUT;
    bool okB = colB < OUT;
    float b1A = okA ? b1[colA] : 0.0f, b1B = okB ? b1[colB] : 0.0f;
    float b2A = okA ? b2[colA] : 0.0f, b2B = okB ? b2[colB] : 0.0f;
    float b3A = okA ? b3[colA] : 0.0f, b3B = okB ? b3[colB] : 0.0f;
    long ss = (long)Nn * OUT;
    int row0 = mt * 16 + half * 8;
#pragma unroll
    for (int r = 0; r < 8; ++r) {
        int row = row0 + r;
        if (row < Nn) {
            long rb = (long)row * OUT;
            if (okA) {
                out[rb + colA]          = acc1A[r] + b1A;
                out[ss + rb + colA]     = acc2A[r] + b2A;
                out[2 * ss + rb + colA] = acc3A[r] + b3A;
            }
            if (okB) {
                out[rb + colB]          = acc1B[r] + b1B;
                out[ss + rb + colB]     = acc2B[r] + b2B;
                out[2 * ss + rb + colB] = acc3B[r] + b3B;
            }
        }
    }
}

// ---------------------------------------------------------------- launch
extern "C" void kernel_launch(void* const* d_in, const int* in_sizes, int n_in,
                              void* d_out, int out_size, void* d_ws, size_t ws_size,
                              hipStream_t stream) {
    const float* x   = (const float*)d_in[0];
    const int*   ei  = (const int*)  d_in[1];
    const float* ew  = (const float*)d_in[2];
    const float* w10 = (const float*)d_in[3];
    const float* b1  = (const float*)d_in[4];
    const float* w20 = (const float*)d_in[5];
    const float* w21 = (const float*)d_in[6];
    const float* b2  = (const float*)d_in[7];
    const float* w30 = (const float*)d_in[8];
    const float* w31 = (const float*)d_in[9];
    const float* w32 = (const float*)d_in[10];
    const float* b3  = (const float*)d_in[11];
    float* out = (float*)d_out;

    const int Nn  = in_sizes[0] / IN_CH;
    const int E   = in_sizes[2];
    const int OUT = in_sizes[3] / IN_CH;
    const int OUT_PAD = ((OUT + 31) / 32) * 32;   // multiple of 32 for NT=2 tiles
    const int* src = ei;
    const int* dst = ei + E;

    // Workspace: pbuf (rows of [tx1 | p2], N*256) | deg | norm | packed weights
    float* pbuf = (float*)d_ws;
    float* deg  = pbuf + (size_t)Nn * 2 * IN_CH;
    float* nrm  = deg + Nn;
    float* wt   = nrm + E;

    long zcount = 2L * Nn * IN_CH + Nn;   // pbuf + deg
    mc_zero_f32<<<2048, 256, 0, stream>>>(pbuf, zcount);

    // Pack/transpose/interleave all six weights (fully overwrites wt)
    int pk_total  = OUT_PAD * NW * IN_CH;
    mc_pack6<<<(pk_total + 255) / 256, 256, 0, stream>>>(w10, w20, w21, w30, w31, w32,
                                                         wt, OUT, OUT_PAD);

    mc_deg<<<(E + 255) / 256, 256, 0, stream>>>(src, ew, deg, E);
    mc_dis<<<(Nn + 255) / 256, 256, 0, stream>>>(deg, Nn);
    mc_norm<<<(E + 255) / 256, 256, 0, stream>>>(src, dst, ew, deg, nrm, E);

    // tx1 = propagate(x)  -> pbuf[row*256 + 0..127]
    mc_prop<<<(E + 7) / 8, 256, 0, stream>>>(x, IN_CH, src, dst, nrm,
                                             pbuf, 2 * IN_CH, E);
    // p2  = propagate(tx1) -> pbuf[row*256 + 128..255]
    mc_prop<<<(E + 7) / 8, 256, 0, stream>>>(pbuf, 2 * IN_CH, src, dst, nrm,
                                             pbuf + IN_CH, 2 * IN_CH, E);

    const int mtiles = (Nn + 15) / 16;
    const int ntp    = OUT_PAD / 32;
    const int tiles  = mtiles * ntp;
    mc_gemm<<<(tiles + 3) / 4, 128, 0, stream>>>(x, pbuf, wt, b1, b2, b3,
                                                 out, Nn, OUT, ntp);
}